// ViSRGCFeatureV1_31602369364616
// MI455X (gfx1250) — compile-verified
//
#include <hip/hip_runtime.h>
#include <math.h>

// Problem constants from the reference
#define B_ 4
#define N_ 128
#define D_ 256
#define K_ 128
#define H_ 32
#define A_ 128

typedef __attribute__((ext_vector_type(2))) float v2f;
typedef __attribute__((ext_vector_type(8))) float v8f;

__device__ __forceinline__ float gelu_exact(float x) {
    // torch exact GELU: 0.5*x*(1+erf(x/sqrt(2)))
    return 0.5f * x * (1.0f + erff(x * 0.70710678118654752f));
}

// ---------------------------------------------------------------------------
// Kernel A: padding mask (keep) per node + no_pos per batch.
// grid = B, block = 128 (one thread per node)
// ---------------------------------------------------------------------------
__global__ void kA(const float* __restrict__ pos, const int* __restrict__ x,
                   float* __restrict__ ws_keep, float* __restrict__ ws_nopos) {
    int b = blockIdx.x;
    int n = threadIdx.x;
    const int* xr = x + (b * N_ + n) * 9;
    int allz = 1;
#pragma unroll
    for (int f = 0; f < 9; ++f) allz &= (xr[f] == 0);
    ws_keep[b * N_ + n] = allz ? 0.0f : 1.0f;

    const float* pr = pos + (b * N_ + n) * 3;
    int nz = (pr[0] != 0.0f) | (pr[1] != 0.0f) | (pr[2] != 0.0f);
    __shared__ int snz;
    if (n == 0) snz = 0;
    __syncthreads();
    if (nz) atomicOr(&snz, 1);
    __syncthreads();
    if (n == 0) ws_nopos[b] = snz ? 1.0f : 0.0f;
}

// ---------------------------------------------------------------------------
// Kernel W: interleave pairs of weight rows so each WMMA B-fragment
// (W[kk][col], W[kk+1][col]) is one aligned 8-byte load.
//   w1p[(k>>1)*512 + col*2 + (k&1)]  = dih_W1[k*256 + col]
//   gw1p[(k>>1)*256 + col*2 + (k&1)] = gbf_W1[k*128 + col]
// ---------------------------------------------------------------------------
__global__ void kW(const float* __restrict__ dih_W1, const float* __restrict__ gbf_W1,
                   float* __restrict__ w1p, float* __restrict__ gw1p) {
    int idx = blockIdx.x * 256 + threadIdx.x;
    if (idx < D_ * D_) {
        int k = idx >> 8, col = idx & 255;
        w1p[(k >> 1) * 512 + col * 2 + (k & 1)] = dih_W1[idx];
    }
    if (idx < K_ * K_) {
        int k = idx >> 7, col = idx & 127;
        gw1p[(k >> 1) * 256 + col * 2 + (k & 1)] = gbf_W1[idx];
    }
}

// ---------------------------------------------------------------------------
// Kernel B: per-node precompute.
//   nf = sum of 9 embedding rows; s[c] = sum_j keep_j * u(i->j)_c
//   l = nf@W_ld, r = nf@W_rd  -> workspace (rank-1 factorization of node_vec)
//   angle = NonLinear(|s|^2 * (nf@W_la) ⊙ (nf@W_ra)) * no_pos -> output
// grid = B*N, block = 256 (one thread per feature d)
// ---------------------------------------------------------------------------
__global__ void __launch_bounds__(256)
kB(const float* __restrict__ pos, const float* __restrict__ atom_emb,
   const float* __restrict__ W_la, const float* __restrict__ W_ra,
   const float* __restrict__ W_ld, const float* __restrict__ W_rd,
   const float* __restrict__ ang_W1, const float* __restrict__ ang_b1,
   const float* __restrict__ ang_W2, const float* __restrict__ ang_b2,
   const int* __restrict__ x, const float* __restrict__ ws_keep,
   const float* __restrict__ ws_nopos,
   float* __restrict__ ws_l, float* __restrict__ ws_r, float* __restrict__ ws_s,
   float* __restrict__ out_angle) {
    const int bi = blockIdx.x;         // b*N + i
    const int b = bi >> 7;
    const int tid = threadIdx.x;       // feature index d

    __shared__ float red0[N_], red1[N_], red2[N_];
    __shared__ float nfs[D_];
    __shared__ float svec[4];
    __shared__ float angs[D_];
    __shared__ float gbuf[D_];

    // node feature (9 gathered embedding rows summed)
    const int* xr = x + bi * 9;
    float nf = 0.0f;
#pragma unroll
    for (int f = 0; f < 9; ++f) {
        int idx = xr[f];
        nf += atom_emb[((size_t)(f * (A_ + 1) + idx)) * D_ + tid];
    }
    nfs[tid] = nf;

    // s[c] = sum_j keep_j * normalized delta(i->j)
    float px = pos[bi * 3 + 0], py = pos[bi * 3 + 1], pz = pos[bi * 3 + 2];
    if (tid < N_) {
        const float* pj = pos + (b * N_ + tid) * 3;
        float dx = pj[0] - px, dy = pj[1] - py, dz = pj[2] - pz;
        float dist = sqrtf(dx * dx + dy * dy + dz * dz);
        float inv = 1.0f / (dist + 1e-5f);
        float kp = ws_keep[b * N_ + tid];
        red0[tid] = kp * dx * inv;
        red1[tid] = kp * dy * inv;
        red2[tid] = kp * dz * inv;
    }
    __syncthreads();
    for (int off = 64; off > 0; off >>= 1) {
        if (tid < off) {
            red0[tid] += red0[tid + off];
            red1[tid] += red1[tid + off];
            red2[tid] += red2[tid + off];
        }
        __syncthreads();
    }
    if (tid == 0) { svec[0] = red0[0]; svec[1] = red1[0]; svec[2] = red2[0]; svec[3] = 0.0f; }
    __syncthreads();
    float sx = svec[0], sy = svec[1], sz = svec[2];
    float ss2 = sx * sx + sy * sy + sz * sz;

    // four matvecs against 256x256 weights (L2-resident)
    float l = 0.0f, r = 0.0f, p = 0.0f, q = 0.0f;
    for (int d = 0; d < D_; ++d) {
        float v = nfs[d];
        l = fmaf(v, W_ld[d * D_ + tid], l);
        r = fmaf(v, W_rd[d * D_ + tid], r);
        p = fmaf(v, W_la[d * D_ + tid], p);
        q = fmaf(v, W_ra[d * D_ + tid], q);
    }
    ws_l[(size_t)bi * D_ + tid] = l;
    ws_r[(size_t)bi * D_ + tid] = r;
    if (tid < 4) ws_s[bi * 4 + tid] = svec[tid];

    // angle branch: angle_in = |s|^2 * p ⊙ q ; two 256x256 matvecs + GELU
    angs[tid] = ss2 * p * q;
    __syncthreads();
    float g = ang_b1[tid];
    for (int d = 0; d < D_; ++d) g = fmaf(angs[d], ang_W1[d * D_ + tid], g);
    gbuf[tid] = gelu_exact(g);
    __syncthreads();
    float o = ang_b2[tid];
    for (int d = 0; d < D_; ++d) o = fmaf(gbuf[d], ang_W2[d * D_ + tid], o);
    out_angle[(size_t)bi * D_ + tid] = o * ws_nopos[b];
}

// ---------------------------------------------------------------------------
// Kernel C: fused pair-tile kernel.  One block handles (b, i, 32 consecutive j).
//   edge_dih[j,d] = coef_j * r_i[d] * l_j[d]                 (rank-1 collapse)
//   edge_feat[j,k] = gaussian(mul*dist+bias; mean_k, std_k)
//   Y1 = GELU(edge_dih @ dih_W1 + b1)   via V_WMMA_F32_16X16X4_F32
//   Y2 = GELU(edge_feat @ gbf_W1 + b1)  via V_WMMA_F32_16X16X4_F32
//   out[j,h] = Y1@dih_W2 + dih_b2 + Y2@gbf_W2 + gbf_b2  (VALU, N=32 skinny)
// Each wave computes a 2x2 grid of 16x16 tiles: B fragments shared across the
// two row tiles, A fragments shared across the two column tiles (4 independent
// accumulator chains -> good XDL pipelining, half the weight traffic).
// grid = B*N*(N/32) = 2048, block = 256 (8 wave32s)
// ---------------------------------------------------------------------------
__global__ void __launch_bounds__(256)
kPair(const float* __restrict__ pos,
      const float* __restrict__ gbf_means, const float* __restrict__ gbf_stds,
      const float* __restrict__ src_mul, const float* __restrict__ tgt_mul,
      const float* __restrict__ src_bias, const float* __restrict__ tgt_bias,
      const float* __restrict__ gbf_b1,
      const float* __restrict__ gbf_W2, const float* __restrict__ gbf_b2,
      const float* __restrict__ dih_b1,
      const float* __restrict__ dih_W2, const float* __restrict__ dih_b2,
      const int* __restrict__ nte,
      const float* __restrict__ ws_l, const float* __restrict__ ws_r,
      const float* __restrict__ ws_s, const float* __restrict__ ws_keep,
      const float* __restrict__ ws_nopos,
      const float* __restrict__ w1p, const float* __restrict__ gw1p,
      float* __restrict__ out_attn, float* __restrict__ out_delta) {
    const int gid = blockIdx.x;
    const int jt = gid & 3;
    const int i = (gid >> 2) & 127;
    const int b = gid >> 9;
    const int tid = threadIdx.x;
    const int lane = tid & 31;
    const int wave = tid >> 5;
    const int j0 = jt * 32;

    __shared__ float A1[32][257];   // edge_dih tile (M=32 pairs x K=256)
    __shared__ float Y1[32][257];   // GELU(GEMM1 dih)
    __shared__ float A2[32][129];   // edge_feature tile (32 x 128)
    __shared__ float Y2[32][129];   // GELU(GEMM1 gbf)
    __shared__ float gmean[K_], gcoef[K_];
    __shared__ float coefs[32], dists[32], mulv[32], biasv[32];
    __shared__ int padj[32];

    if (tid < K_) {
        gmean[tid] = gbf_means[tid];
        gcoef[tid] = 1.0f / (fabsf(gbf_stds[tid]) + 0.01f);
    }

    // per-pair scalars + delta_pos output
    if (tid < 32) {
        const int m = tid, j = j0 + m;
        float pix = pos[(b * N_ + i) * 3 + 0], piy = pos[(b * N_ + i) * 3 + 1],
              piz = pos[(b * N_ + i) * 3 + 2];
        float pjx = pos[(b * N_ + j) * 3 + 0], pjy = pos[(b * N_ + j) * 3 + 1],
              pjz = pos[(b * N_ + j) * 3 + 2];
        float dx = pjx - pix, dy = pjy - piy, dz = pjz - piz;
        float dist = sqrtf(dx * dx + dy * dy + dz * dz);
        float inv = 1.0f / (dist + 1e-5f);
        float ux = dx * inv, uy = dy * inv, uz = dz * inv;
        float uu = ux * ux + uy * uy + uz * uz;
        float six = ws_s[(b * N_ + i) * 4 + 0], siy = ws_s[(b * N_ + i) * 4 + 1],
              siz = ws_s[(b * N_ + i) * 4 + 2];
        float sjx = ws_s[(b * N_ + j) * 4 + 0], sjy = ws_s[(b * N_ + j) * 4 + 1],
              sjz = ws_s[(b * N_ + j) * 4 + 2];
        float sisj = six * sjx + siy * sjy + siz * sjz;
        float sju = sjx * ux + sjy * uy + sjz * uz;
        float siu = six * ux + siy * uy + siz * uz;
        coefs[m] = sisj - sju * siu * (2.0f - uu);
        dists[m] = dist;
        int e0 = nte[(((size_t)(b * N_ + i)) * N_ + j) * 2 + 0];
        int e1 = nte[(((size_t)(b * N_ + i)) * N_ + j) * 2 + 1];
        mulv[m] = src_mul[e0] + tgt_mul[e1];
        biasv[m] = src_bias[e0] + tgt_bias[e1];
        padj[m] = (ws_keep[b * N_ + j] == 0.0f) ? 1 : 0;
        float* dp = out_delta + (((size_t)(b * N_ + i)) * N_ + j) * 3;
        dp[0] = ux; dp[1] = uy; dp[2] = uz;
    }
    __syncthreads();

    // build A1 (edge_dih): coef_j * r_i[d] * l_j[d]
    float rv = ws_r[((size_t)(b * N_ + i)) * D_ + tid];
#pragma unroll 4
    for (int m = 0; m < 32; ++m) {
        float lv = ws_l[((size_t)(b * N_ + j0 + m)) * D_ + tid];
        A1[m][tid] = coefs[m] * rv * lv;
    }
    // build A2 (edge_feature gaussians)
    if (tid < K_) {
        float gm = gmean[tid], gc = gcoef[tid];
#pragma unroll 4
        for (int m = 0; m < 32; ++m) {
            float xg = fmaf(mulv[m], dists[m], biasv[m]);
            float t = (xg - gm) * gc;
            A2[m][tid] = expf(-0.5f * t * t) * gc * 0.39894263f;  // 1/sqrt(2*3.14159)
        }
    }
    __syncthreads();

    const int row = lane & 15;   // A: M index ; B/D: N index
    const int kh = lane >> 4;    // K half selector (ISA 16x4 f32 A/B layout)

    // ---- GEMM1 dihedral: [32x256] @ dih_W1[256x256] via f32 WMMA ----
    {
        const int colA = (wave * 2 + 0) * 16 + row;
        const int colB = (wave * 2 + 1) * 16 + row;
        const v2f* w1v = (const v2f*)w1p;   // pair-of-rows interleaved
        v8f acc00 = {0.f,0.f,0.f,0.f,0.f,0.f,0.f,0.f};
        v8f acc01 = acc00, acc10 = acc00, acc11 = acc00;
        for (int k0 = 0; k0 < D_; k0 += 4) {
            const int kk = k0 + kh * 2;
            const int krow = (k0 >> 1) + kh;
            v2f a0; a0.x = A1[row][kk];      a0.y = A1[row][kk + 1];
            v2f a1; a1.x = A1[16 + row][kk]; a1.y = A1[16 + row][kk + 1];
            v2f b0 = w1v[krow * 256 + colA];
            v2f b1 = w1v[krow * 256 + colB];
            acc00 = __builtin_amdgcn_wmma_f32_16x16x4_f32(false, a0, false, b0, (short)0, acc00, false, false);
            acc10 = __builtin_amdgcn_wmma_f32_16x16x4_f32(false, a1, false, b0, (short)0, acc10, false, false);
            acc01 = __builtin_amdgcn_wmma_f32_16x16x4_f32(false, a0, false, b1, (short)0, acc01, false, false);
            acc11 = __builtin_amdgcn_wmma_f32_16x16x4_f32(false, a1, false, b1, (short)0, acc11, false, false);
        }
        const float biasA = dih_b1[colA], biasB = dih_b1[colB];
#pragma unroll
        for (int v = 0; v < 8; ++v) {
            const int mr = v + 8 * kh;
            Y1[mr][colA]      = gelu_exact(acc00[v] + biasA);
            Y1[16 + mr][colA] = gelu_exact(acc10[v] + biasA);
            Y1[mr][colB]      = gelu_exact(acc01[v] + biasB);
            Y1[16 + mr][colB] = gelu_exact(acc11[v] + biasB);
        }
    }
    // ---- GEMM1 gbf: [32x128] @ gbf_W1[128x128] via f32 WMMA ----
    {
        const int col = wave * 16 + row;  // 8 waves cover 128 cols
        const v2f* gwv = (const v2f*)gw1p;
        v8f acc0 = {0.f,0.f,0.f,0.f,0.f,0.f,0.f,0.f};
        v8f acc1 = acc0;
        for (int k0 = 0; k0 < K_; k0 += 4) {
            const int kk = k0 + kh * 2;
            const int krow = (k0 >> 1) + kh;
            v2f a0; a0.x = A2[row][kk];      a0.y = A2[row][kk + 1];
            v2f a1; a1.x = A2[16 + row][kk]; a1.y = A2[16 + row][kk + 1];
            v2f bb = gwv[krow * 128 + col];
            acc0 = __builtin_amdgcn_wmma_f32_16x16x4_f32(false, a0, false, bb, (short)0, acc0, false, false);
            acc1 = __builtin_amdgcn_wmma_f32_16x16x4_f32(false, a1, false, bb, (short)0, acc1, false, false);
        }
        const float bias = gbf_b1[col];
#pragma unroll
        for (int v = 0; v < 8; ++v) {
            const int mr = v + 8 * kh;
            Y2[mr][col]      = gelu_exact(acc0[v] + bias);
            Y2[16 + mr][col] = gelu_exact(acc1[v] + bias);
        }
    }
    __syncthreads();

    // ---- GEMM2 (skinny N=32): out[m,h] = Y1@dih_W2 + Y2@gbf_W2 + biases ----
    {
        const int m = tid >> 3;           // 0..31 (pair row)
        const int h0 = (tid & 7) * 4;     // 4 consecutive heads
        const int hq = h0 >> 2;
        float a0 = dih_b2[h0 + 0] + gbf_b2[h0 + 0];
        float a1 = dih_b2[h0 + 1] + gbf_b2[h0 + 1];
        float a2 = dih_b2[h0 + 2] + gbf_b2[h0 + 2];
        float a3 = dih_b2[h0 + 3] + gbf_b2[h0 + 3];
        const float4* w2d = (const float4*)dih_W2;
        const float4* w2g = (const float4*)gbf_W2;
        for (int d = 0; d < D_; ++d) {
            float y = Y1[m][d];
            float4 w = w2d[d * 8 + hq];
            a0 = fmaf(y, w.x, a0); a1 = fmaf(y, w.y, a1);
            a2 = fmaf(y, w.z, a2); a3 = fmaf(y, w.w, a3);
        }
        for (int k = 0; k < K_; ++k) {
            float y = Y2[m][k];
            float4 w = w2g[k * 8 + hq];
            a0 = fmaf(y, w.x, a0); a1 = fmaf(y, w.y, a1);
            a2 = fmaf(y, w.z, a2); a3 = fmaf(y, w.w, a3);
        }
        const float np = ws_nopos[b];
        a0 *= np; a1 *= np; a2 *= np; a3 *= np;
        if (padj[m]) {
            a0 = -__builtin_inff(); a1 = -__builtin_inff();
            a2 = -__builtin_inff(); a3 = -__builtin_inff();
        }
        const int j = j0 + m;
        out_attn[(((size_t)b * H_ + h0 + 0) * N_ + i) * N_ + j] = a0;
        out_attn[(((size_t)b * H_ + h0 + 1) * N_ + i) * N_ + j] = a1;
        out_attn[(((size_t)b * H_ + h0 + 2) * N_ + i) * N_ + j] = a2;
        out_attn[(((size_t)b * H_ + h0 + 3) * N_ + i) * N_ + j] = a3;
    }
}

// ---------------------------------------------------------------------------
extern "C" void kernel_launch(void* const* d_in, const int* in_sizes, int n_in,
                              void* d_out, int out_size, void* d_ws, size_t ws_size,
                              hipStream_t stream) {
    (void)in_sizes; (void)n_in; (void)out_size; (void)ws_size;
    const float* pos       = (const float*)d_in[0];
    const float* atom_emb  = (const float*)d_in[1];
    const float* W_la      = (const float*)d_in[2];
    const float* W_ra      = (const float*)d_in[3];
    const float* W_ld      = (const float*)d_in[4];
    const float* W_rd      = (const float*)d_in[5];
    const float* gbf_means = (const float*)d_in[6];
    const float* gbf_stds  = (const float*)d_in[7];
    const float* src_mul   = (const float*)d_in[8];
    const float* tgt_mul   = (const float*)d_in[9];
    const float* src_bias  = (const float*)d_in[10];
    const float* tgt_bias  = (const float*)d_in[11];
    const float* gbf_W1    = (const float*)d_in[12];
    const float* gbf_b1    = (const float*)d_in[13];
    const float* gbf_W2    = (const float*)d_in[14];
    const float* gbf_b2    = (const float*)d_in[15];
    const float* dih_W1    = (const float*)d_in[16];
    const float* dih_b1    = (const float*)d_in[17];
    const float* dih_W2    = (const float*)d_in[18];
    const float* dih_b2    = (const float*)d_in[19];
    const float* ang_W1    = (const float*)d_in[20];
    const float* ang_b1    = (const float*)d_in[21];
    const float* ang_W2    = (const float*)d_in[22];
    const float* ang_b2    = (const float*)d_in[23];
    const int*   x         = (const int*)d_in[24];
    const int*   nte       = (const int*)d_in[25];

    // workspace layout (floats):
    //   l[131072] r[131072] s[2048] keep[512] nopos[64 pad]
    //   w1p[65536] gw1p[16384]
    float* ws      = (float*)d_ws;
    float* ws_l    = ws;
    float* ws_r    = ws + 131072;
    float* ws_s    = ws + 262144;
    float* ws_keep = ws + 264192;
    float* ws_np   = ws + 264704;
    float* ws_w1p  = ws + 264768;
    float* ws_gw1p = ws + 330304;

    // outputs: attn_bias[4,32,128,128] | angle[4,128,256] | delta_pos[4,128,128,3]
    float* out       = (float*)d_out;
    float* out_attn  = out;
    float* out_angle = out + 2097152;
    float* out_delta = out + 2228224;

    kA<<<B_, 128, 0, stream>>>(pos, x, ws_keep, ws_np);
    kW<<<(D_ * D_) / 256, 256, 0, stream>>>(dih_W1, gbf_W1, ws_w1p, ws_gw1p);
    kB<<<B_ * N_, 256, 0, stream>>>(pos, atom_emb, W_la, W_ra, W_ld, W_rd,
                                    ang_W1, ang_b1, ang_W2, ang_b2,
                                    x, ws_keep, ws_np,
                                    ws_l, ws_r, ws_s, out_angle);
    kPair<<<B_ * N_ * (N_ / 32), 256, 0, stream>>>(
        pos, gbf_means, gbf_stds, src_mul, tgt_mul, src_bias, tgt_bias,
        gbf_b1, gbf_W2, gbf_b2, dih_b1, dih_W2, dih_b2,
        nte, ws_l, ws_r, ws_s, ws_keep, ws_np, ws_w1p, ws_gw1p,
        out_attn, out_delta);
}